// DefaultOClusterSegmentor_2508260901472
// MI455X (gfx1250) — compile-verified
//
#include <hip/hip_runtime.h>
#include <stdint.h>

typedef float v2f __attribute__((ext_vector_type(2)));
typedef float v8f __attribute__((ext_vector_type(8)));

#define THREADS 256
#define SORT_BLOCK 1024
#define SORT_CHUNK 2048
#define I64MAX 0x7fffffffffffffffLL
#define BIGF 1e38f

__device__ __forceinline__ long long fnv5(long long b, long long c, long long x,
                                          long long y, long long z) {
  unsigned long long h = 0xCBF29CE484222325ULL;  // FNV_OFF bits
  h = (h ^ (unsigned long long)b) * 4294967731ULL;
  h = (h ^ (unsigned long long)c) * 4294967731ULL;
  h = (h ^ (unsigned long long)x) * 4294967731ULL;
  h = (h ^ (unsigned long long)y) * 4294967731ULL;
  h = (h ^ (unsigned long long)z) * 4294967731ULL;
  return (long long)h;
}

__device__ __forceinline__ int fsign(float x) { return (x > 0.f) - (x < 0.f); }

// ---------------- K0: per-point keys ----------------
__global__ void k_prep(const float* grid, const long long* batch, const long long* label,
                       const int* pBase, const int* pNcls,
                       long long* key1, int* idx1, int* keyBL, int N, int Np2) {
  int i = blockIdx.x * blockDim.x + threadIdx.x;
  if (i >= Np2) return;
  if (i >= N) { key1[i] = I64MAX; idx1[i] = i; return; }
  float fb = (float)(*pBase);
  long long vx = (long long)floorf(grid[3*i+0] / fb);
  long long vy = (long long)floorf(grid[3*i+1] / fb);
  long long vz = (long long)floorf(grid[3*i+2] / fb);
  long long b = batch[i];
  key1[i] = ((b * 1024 + vx) * 1024 + vy) * 1024 + vz;
  idx1[i] = i;
  keyBL[i] = (int)(b * (long long)(*pNcls) + label[i]);
}

// ---------------- bitonic sort: global step ----------------
__global__ void k_bitonic(long long* key, int* idx, int j, int k, int n) {
  int i = blockIdx.x * blockDim.x + threadIdx.x;
  int p = i ^ j;
  if (i >= n || p <= i || p >= n) return;
  long long ki = key[i], kp = key[p];
  bool up = ((i & k) == 0);
  if (up ? (ki > kp) : (ki < kp)) {
    key[i] = kp; key[p] = ki;
    int t = idx[i]; idx[i] = idx[p]; idx[p] = t;
  }
}

// ---------------- bitonic sort: LDS-fused local full sort (stages k<=SORT_CHUNK) ----
__global__ void k_sort_local(long long* key, int* idx, int n) {
  __shared__ long long sk[SORT_CHUNK];
  __shared__ int si[SORT_CHUNK];
  int t = threadIdx.x;
  int base = blockIdx.x * SORT_CHUNK;
  for (int e = t; e < SORT_CHUNK; e += SORT_BLOCK) {
    int g = base + e;
    sk[e] = (g < n) ? key[g] : I64MAX;
    si[e] = (g < n) ? idx[g] : g;
  }
  __syncthreads();
  for (int k = 2; k <= SORT_CHUNK; k <<= 1) {
    for (int j = k >> 1; j > 0; j >>= 1) {
      int i = ((t & ~(j - 1)) << 1) | (t & (j - 1));
      int p = i | j;
      bool up = (((base + i) & k) == 0);
      long long ki = sk[i], kp = sk[p];
      if (up ? (ki > kp) : (ki < kp)) {
        sk[i] = kp; sk[p] = ki;
        int tt = si[i]; si[i] = si[p]; si[p] = tt;
      }
      __syncthreads();
    }
  }
  for (int e = t; e < SORT_CHUNK; e += SORT_BLOCK) {
    int g = base + e;
    if (g < n) { key[g] = sk[e]; idx[g] = si[e]; }
  }
}

// ---------------- bitonic sort: LDS-fused local merge (passes j<=SORT_CHUNK/2) ------
__global__ void k_merge_local(long long* key, int* idx, int K, int n) {
  __shared__ long long sk[SORT_CHUNK];
  __shared__ int si[SORT_CHUNK];
  int t = threadIdx.x;
  int base = blockIdx.x * SORT_CHUNK;
  for (int e = t; e < SORT_CHUNK; e += SORT_BLOCK) {
    int g = base + e;
    sk[e] = (g < n) ? key[g] : I64MAX;
    si[e] = (g < n) ? idx[g] : g;
  }
  __syncthreads();
  for (int j = SORT_CHUNK >> 1; j > 0; j >>= 1) {
    int i = ((t & ~(j - 1)) << 1) | (t & (j - 1));
    int p = i | j;
    bool up = (((base + i) & K) == 0);
    long long ki = sk[i], kp = sk[p];
    if (up ? (ki > kp) : (ki < kp)) {
      sk[i] = kp; sk[p] = ki;
      int tt = si[i]; si[i] = si[p]; si[p] = tt;
    }
    __syncthreads();
  }
  for (int e = t; e < SORT_CHUNK; e += SORT_BLOCK) {
    int g = base + e;
    if (g < n) { key[g] = sk[e]; idx[g] = si[e]; }
  }
}

// ---------------- K2: boundary flags -> cluster ids (single block, deterministic) ----
__global__ void k_scan(const long long* skey, const int* sidx, int* clOfPt,
                       int* clStart, int* hdr, int N) {
  __shared__ int sh[THREADS];
  __shared__ int offset;
  int t = threadIdx.x;
  if (t == 0) offset = 0;
  __syncthreads();
  for (int base = 0; base < N; base += THREADS) {
    int i = base + t;
    int flag = 0;
    if (i < N) flag = (i == 0) || (skey[i] != skey[i-1]);
    sh[t] = flag;
    __syncthreads();
    for (int d = 1; d < THREADS; d <<= 1) {
      int v = (t >= d) ? sh[t-d] : 0;
      __syncthreads();
      sh[t] += v;
      __syncthreads();
    }
    if (i < N) {
      int cid = offset + sh[t] - 1;
      clOfPt[sidx[i]] = cid;
      if (flag) clStart[cid] = i;
    }
    __syncthreads();
    if (t == 0) offset += sh[THREADS-1];
    __syncthreads();
  }
  if (t == 0) { hdr[0] = offset; clStart[offset] = N; }
}

// ---------------- K4: per-cluster stats ----------------
__global__ void k_stats(const float* grid, const long long* batch, const long long* label,
                        const int* sidx, const int* clStart, const int* hdr,
                        const int* pBase, const int* pNcls,
                        float* cxA, float* cyA, float* czA, float4* bmat,
                        int* code, int* pure, int* vxA, int* vyA, int* vzA,
                        long long* key2, int* idx2, int N, int Np2) {
  int c = blockIdx.x * blockDim.x + threadIdx.x;
  if (c >= Np2) return;
  int C = hdr[0];
  if (c >= C) {
    key2[c] = I64MAX;
    idx2[c] = (c < N) ? c : 0;
    if (c < N) {
      pure[c] = 0; code[c] = -1;
      bmat[c] = make_float4(0.f, 0.f, 0.f, 0.f);
    }
    return;
  }
  int s = clStart[c], e = clStart[c+1];
  int p0 = sidx[s];
  float fb = (float)(*pBase);
  long long vx = (long long)floorf(grid[3*p0+0] / fb);
  long long vy = (long long)floorf(grid[3*p0+1] / fb);
  long long vz = (long long)floorf(grid[3*p0+2] / fb);
  long long b = batch[p0];
  double sx = 0.0, sy = 0.0, sz = 0.0;
  long long lmin = I64MAX, lmax = (-I64MAX - 1LL);
  for (int pos = s; pos < e; ++pos) {
    int p = sidx[pos];
    sx += (double)grid[3*p+0]; sy += (double)grid[3*p+1]; sz += (double)grid[3*p+2];
    long long l = label[p];
    if (l < lmin) lmin = l;
    if (l > lmax) lmax = l;
  }
  double cnt = (double)(e - s);
  float fx = (float)(sx / cnt), fy = (float)(sy / cnt), fz = (float)(sz / cnt);
  cxA[c] = fx; cyA[c] = fy; czA[c] = fz;
  bmat[c] = make_float4(-2.f*fx, -2.f*fy, -2.f*fz, fx*fx + fy*fy + fz*fz);
  int pu = (lmin == lmax) ? 1 : 0;
  pure[c] = pu;
  vxA[c] = (int)vx; vyA[c] = (int)vy; vzA[c] = (int)vz;
  code[c] = pu ? (int)(b * (long long)(*pNcls) + lmin) : -1;
  key2[c] = pu ? fnv5(b, lmin, vx, vy, vz) : I64MAX;
  idx2[c] = c;
}

// ---------------- cluster-code sort keys ----------------
__global__ void k_ckey(const int* code, const int* hdr, long long* key3, int* idx3,
                       int N, int Np2) {
  int c = blockIdx.x * blockDim.x + threadIdx.x;
  if (c >= Np2) return;
  int C = hdr[0];
  long long hi;
  if (c < C) {
    int k = code[c];
    hi = (k < 0) ? 0x7fffffffLL : (long long)k;
  } else {
    hi = 0x7fffffffLL;
  }
  key3[c] = (hi << 32) | (unsigned int)c;
  idx3[c] = (c < N) ? c : (N - 1);
}

// ---------------- gather code-sorted cluster data ----------------
__global__ void k_gather(const int* idx3, const int* code, const float4* bmat,
                         int* scode, float4* sbmat, int N) {
  int s = blockIdx.x * blockDim.x + threadIdx.x;
  if (s >= N) return;
  int c = idx3[s];
  scode[s] = code[c];
  sbmat[s] = bmat[c];
}

// ---------------- column-range table: init / boundaries / fill ----------------
__global__ void k_initcol(int* colStart) {
  int t = threadIdx.x;
  if (t < 128) colStart[t] = -1;
}
__global__ void k_bound(const int* scode, int* colStart,
                        const int* pNcls, const int* pNb, int N) {
  int s = blockIdx.x * blockDim.x + threadIdx.x;
  if (s >= N) return;
  int cur = scode[s];
  if (s == 0 || scode[s-1] != cur) {
    int nbl = (*pNcls) * (*pNb);
    if (cur >= 0) colStart[cur] = s;
    else colStart[nbl] = s;
  }
}
__global__ void k_fill(int* colStart, const int* hdr, const int* pNcls, const int* pNb) {
  int nbl = (*pNcls) * (*pNb);
  if (colStart[nbl] < 0) colStart[nbl] = hdr[0];
  for (int k = nbl - 1; k >= 0; --k)
    if (colStart[k] < 0) colStart[k] = colStart[k + 1];
}

// ---------------- point-code sort keys (reuses key1/idx1 after k_stats) ------------
__global__ void k_pkey(const int* keyBL, long long* key1, int* idx1, int N, int Np2) {
  int i = blockIdx.x * blockDim.x + threadIdx.x;
  if (i >= Np2) return;
  key1[i] = (i < N) ? (((long long)keyBL[i] << 32) | (unsigned int)i) : I64MAX;
  idx1[i] = (i < N) ? i : 0;
}

// ---------------- K6: per-(batch,label) centers, fixed-order reduction --------------
__global__ void k_glob(const float* grid, const int* keyBL, double* gc,
                       const int* pNcls, const int* pNb, int N) {
  int bl = blockIdx.x;
  int nbl = (*pNcls) * (*pNb);
  if (bl >= nbl) return;
  __shared__ double sc[THREADS], sx[THREADS], sy[THREADS], sz[THREADS];
  int t = threadIdx.x;
  double c = 0, x = 0, y = 0, z = 0;
  for (int i = t; i < N; i += THREADS) {
    if (keyBL[i] == bl) {
      c += 1.0; x += (double)grid[3*i]; y += (double)grid[3*i+1]; z += (double)grid[3*i+2];
    }
  }
  sc[t] = c; sx[t] = x; sy[t] = y; sz[t] = z;
  __syncthreads();
  for (int s = THREADS/2; s > 0; s >>= 1) {
    if (t < s) { sc[t]+=sc[t+s]; sx[t]+=sx[t+s]; sy[t]+=sy[t+s]; sz[t]+=sz[t+s]; }
    __syncthreads();
  }
  if (t == 0) { gc[4*bl] = sc[0]; gc[4*bl+1] = sx[0]; gc[4*bl+2] = sy[0]; gc[4*bl+3] = sz[0]; }
}

// ---------------- K7: +/-1, +/-2 voxel hash probe ----------------
__global__ void k_probe(const float* grid, const long long* batch, const long long* label,
                        const int* clOfPt, const int* keyBL,
                        const float* cxA, const float* cyA, const float* czA,
                        const int* pure, const int* vxA, const int* vyA, const int* vzA,
                        const long long* key2, const int* idx2, const double* gc,
                        float* tx, float* ty, float* tz, int* missA, int N) {
  int i = blockIdx.x * blockDim.x + threadIdx.x;
  if (i >= N) return;
  int c = clOfPt[i];
  float gx = grid[3*i], gy = grid[3*i+1], gz = grid[3*i+2];
  int kbl = keyBL[i];
  double cnt = gc[4*kbl];
  double dn = cnt > 1.0 ? cnt : 1.0;
  float gcx = (float)(gc[4*kbl+1] / dn);
  float gcy = (float)(gc[4*kbl+2] / dn);
  float gcz = (float)(gc[4*kbl+3] / dn);
  int sxs = fsign(gcx - cxA[c]);
  int sys = fsign(gcy - cyA[c]);
  int szs = fsign(gcz - czA[c]);
  long long b = batch[i], l = label[i];
  float ox = gx, oy = gy, oz = gz;
  int hit = 0;
  for (int step = 1; step <= 2 && !hit; ++step) {
    long long cvx = (long long)vxA[c] + (long long)(sxs * step);
    long long cvy = (long long)vyA[c] + (long long)(sys * step);
    long long cvz = (long long)vzA[c] + (long long)(szs * step);
    long long ck = fnv5(b, l, cvx, cvy, cvz);
    int lo = 0, hi = N;
    while (lo < hi) { int m = (lo + hi) >> 1; if (key2[m] < ck) lo = m + 1; else hi = m; }
    int ic = lo < N - 1 ? lo : N - 1;
    if (lo < N && key2[ic] == ck && pure[idx2[ic]]) {
      int cc = idx2[ic];
      ox = cxA[cc]; oy = cyA[cc]; oz = czA[cc];
      hit = 1;
    }
  }
  tx[i] = ox; ty[i] = oy; tz[i] = oz;
  missA[i] = (!pure[c]) && (!hit);
}

// ---------------- K8: code-grouped WMMA masked nearest-center search ----------------
// S[i][j] = -2 g_i . c_j + |c_j|^2 via V_WMMA_F32_16X16X4_F32 (row-const |g|^2 dropped:
// argmin-invariant). Rows & columns sorted by (batch,label) code, so each 16-row wave
// only sweeps the matching column range; per-cell code mask keeps boundaries exact.
__global__ void k_fallback(const float* grid, const int* keyBL, const int* rowPt,
                           const float4* sbmat, const int* scode, const int* sclid,
                           const int* colStart, int* bestJ, int N) {
  int lane = threadIdx.x & 31;
  int wv = blockIdx.x * (blockDim.x >> 5) + (threadIdx.x >> 5);
  int rowBase = wv * 16;
  if (rowBase >= N) return;
  int col = lane & 15;
  int half = lane >> 4;

  // A operand per documented 16x4 f32 layout: lanes 0-15 hold (K0,K1), 16-31 (K2,K3)
  int rp = rowPt[rowBase + col];
  v2f a;
  if (half == 0) { a.x = grid[3*rp+0]; a.y = grid[3*rp+1]; }
  else           { a.x = grid[3*rp+2]; a.y = 1.0f; }

  int codey[8];
  int cmin = 0x7fffffff, cmax = 0;
#pragma unroll
  for (int v = 0; v < 8; ++v) {
    int row = rowBase + v + 8*half;
    int cy = keyBL[rowPt[row]];
    codey[v] = cy;
    cmin = cy < cmin ? cy : cmin;
    cmax = cy > cmax ? cy : cmax;
  }
#pragma unroll
  for (int off = 16; off >= 1; off >>= 1) {
    int om = __shfl_xor(cmin, off, 32);
    int ox = __shfl_xor(cmax, off, 32);
    cmin = om < cmin ? om : cmin;
    cmax = ox > cmax ? ox : cmax;
  }
  int jStart = colStart[cmin];
  int jEnd   = colStart[cmax + 1];
  int t0 = jStart >> 4;
  int t1 = (jEnd + 15) >> 4;

  float bv[8]; int bj[8];
#pragma unroll
  for (int v = 0; v < 8; ++v) { bv[v] = BIGF; bj[v] = -1; }

  const v2f* bb = (const v2f*)sbmat;
  for (int t = t0; t < t1; ++t) {
    int j = t * 16 + col;
    v2f b = bb[2*j + half];   // unconditional b64 load (padded deterministically)
    int cxj = scode[j];
    v8f acc = {};
    acc = __builtin_amdgcn_wmma_f32_16x16x4_f32(false, a, false, b, (short)0, acc,
                                                false, false);
#pragma unroll
    for (int v = 0; v < 8; ++v) {
      float val = acc[v];     // cell (M = v + 8*half, Ncol = col)
      if (cxj == codey[v] && val < bv[v]) { bv[v] = val; bj[v] = j; }
    }
  }
  // argmin across the 16 lanes of each half (tie-break: smallest sorted position
  // = smallest cluster id within the code group, matching the reference)
#pragma unroll
  for (int v = 0; v < 8; ++v) {
#pragma unroll
    for (int off = 8; off >= 1; off >>= 1) {
      float ov = __shfl_xor(bv[v], off, 32);
      int   oj = __shfl_xor(bj[v], off, 32);
      if (ov < bv[v] || (ov == bv[v] && (unsigned)oj < (unsigned)bj[v])) {
        bv[v] = ov; bj[v] = oj;
      }
    }
  }
  if (col == 0) {
#pragma unroll
    for (int v = 0; v < 8; ++v) {
      int row = rowBase + v + 8*half;
      int rp2 = rowPt[row];
      bestJ[rp2] = (bj[v] >= 0) ? sclid[bj[v]] : -1;
    }
  }
}

// ---------------- K9: apply fallback, compute offsets + mag, fill quantile keys ------
__global__ void k_apply(const float* grid, const int* missA, const int* bestJ,
                        const float* cxA, const float* cyA, const float* czA,
                        float* tx, float* ty, float* tz, float* mag,
                        long long* key1, int N, int Np2) {
  int i = blockIdx.x * blockDim.x + threadIdx.x;
  if (i >= Np2) return;
  if (i >= N) { key1[i] = I64MAX; return; }
  float ox = tx[i], oy = ty[i], oz = tz[i];
  int bj = bestJ[i];
  if (missA[i] && bj >= 0) { ox = cxA[bj]; oy = cyA[bj]; oz = czA[bj]; }
  ox -= grid[3*i]; oy -= grid[3*i+1]; oz -= grid[3*i+2];
  tx[i] = ox; ty[i] = oy; tz[i] = oz;
  float s = ox*ox + oy*oy + oz*oz;
  float m = (s > 0.f) ? sqrtf(s) : 0.f;
  mag[i] = m;
  key1[i] = (long long)(unsigned long long)__float_as_uint(m);
}

// ---------------- K10: final losses, fixed-order reduction ----------------
__global__ void k_loss(const float* pred, const float* tx, const float* ty, const float* tz,
                       const float* mag, const long long* skey, float* out, int N) {
  __shared__ double s0[THREADS], s1[THREADS], s2[THREADS], s3[THREADS];
  int t = threadIdx.x;
  double q = 0.99 * (double)(N - 1);
  int lo = (int)q;
  double fr = q - (double)lo;
  int hi2 = (lo + 1 < N) ? lo + 1 : N - 1;
  float qa = __uint_as_float((unsigned)skey[lo]);
  float qb = __uint_as_float((unsigned)skey[hi2]);
  float thresh = (float)((1.0 - fr) * (double)qa + fr * (double)qb);

  double shub = 0, sn1 = 0, scos = 0, snmd = 0;
  for (int i = t; i < N; i += THREADS) {
    float m = mag[i];
    if (m <= thresh) {
      float px = pred[3*i], py = pred[3*i+1], pz = pred[3*i+2];
      float ox = tx[i], oy = ty[i], oz = tz[i];
      float dx = px - ox, dy = py - oy, dz = pz - oz;
      float ax = fabsf(dx), ay = fabsf(dy), az = fabsf(dz);
      float h = (ax < 1.f ? 0.5f*dx*dx : ax - 0.5f)
              + (ay < 1.f ? 0.5f*dy*dy : ay - 0.5f)
              + (az < 1.f ? 0.5f*dz*dz : az - 0.5f);
      shub += (double)h; sn1 += 1.0;
      if (m > 0.f) {
        float ps = px*px + py*py + pz*pz;
        float pn = (ps > 0.f) ? sqrtf(ps) : 0.f;
        float cosv = (px*ox + py*oy + pz*oz) / fmaxf(pn * m, 1e-4f);
        scos += (double)cosv; snmd += 1.0;
      }
    }
  }
  s0[t] = shub; s1[t] = sn1; s2[t] = scos; s3[t] = snmd;
  __syncthreads();
  for (int s = THREADS/2; s > 0; s >>= 1) {
    if (t < s) { s0[t]+=s0[t+s]; s1[t]+=s1[t+s]; s2[t]+=s2[t+s]; s3[t]+=s3[t+s]; }
    __syncthreads();
  }
  if (t == 0) {
    double n1 = s1[0], nmd = s3[0];
    out[0] = (n1 > 0.0) ? (float)(s0[0] / fmax(n1 * 3.0, 1.0)) : 0.f;
    out[1] = (nmd > 0.0) ? (float)(1.0 - s2[0] / fmax(nmd, 1.0)) : 0.f;
  }
}

// ---------------- host: fused-bitonic driver (21 launches for 64K vs 136) ------------
static void sort_pairs(long long* key, int* idx, int n, hipStream_t s) {
  int gb = (n + THREADS - 1) / THREADS;
  if (n >= SORT_CHUNK) {
    int nb = n / SORT_CHUNK;
    k_sort_local<<<nb, SORT_BLOCK, 0, s>>>(key, idx, n);
    for (int k = SORT_CHUNK * 2; k <= n; k <<= 1) {
      for (int j = k >> 1; j >= SORT_CHUNK; j >>= 1)
        k_bitonic<<<gb, THREADS, 0, s>>>(key, idx, j, k, n);
      k_merge_local<<<nb, SORT_BLOCK, 0, s>>>(key, idx, k, n);
    }
  } else {
    for (int k = 2; k <= n; k <<= 1)
      for (int j = k >> 1; j > 0; j >>= 1)
        k_bitonic<<<gb, THREADS, 0, s>>>(key, idx, j, k, n);
  }
}

extern "C" void kernel_launch(void* const* d_in, const int* in_sizes, int n_in,
                              void* d_out, int out_size, void* d_ws, size_t ws_size,
                              hipStream_t stream) {
  const float*     pred  = (const float*)d_in[0];
  const float*     grid  = (const float*)d_in[1];
  const long long* label = (const long long*)d_in[2];
  const long long* batch = (const long long*)d_in[3];
  const int* pBase = (const int*)d_in[4];   // small positive ints: first 4 LE bytes valid
  const int* pNcls = (const int*)d_in[5];
  const int* pNb   = (const int*)d_in[6];

  int N = in_sizes[0] / 3;
  int Np2 = 1; while (Np2 < N) Np2 <<= 1;

  char* w = (char*)d_ws;
  size_t off = 0;
  auto alloc = [&](size_t bytes) -> char* {
    char* p = w + off; off += (bytes + 255) & ~(size_t)255; return p;
  };
  int*       hdr      = (int*)      alloc(256);
  int*       colStart = (int*)      alloc(512);
  long long* key1     = (long long*)alloc((size_t)Np2 * 8);
  int*       idx1     = (int*)      alloc((size_t)Np2 * 4);
  long long* key2     = (long long*)alloc((size_t)Np2 * 8);
  int*       idx2     = (int*)      alloc((size_t)Np2 * 4);
  long long* key3     = (long long*)alloc((size_t)Np2 * 8);
  int*       idx3     = (int*)      alloc((size_t)Np2 * 4);
  int*       clOfPt   = (int*)      alloc((size_t)N * 4);
  int*       clStart  = (int*)      alloc(((size_t)N + 1) * 4);
  float*     cxA      = (float*)    alloc((size_t)N * 4);
  float*     cyA      = (float*)    alloc((size_t)N * 4);
  float*     czA      = (float*)    alloc((size_t)N * 4);
  float4*    bmat     = (float4*)   alloc((size_t)N * 16);
  float4*    sbmat    = (float4*)   alloc((size_t)N * 16);
  int*       code     = (int*)      alloc((size_t)N * 4);
  int*       scode    = (int*)      alloc((size_t)N * 4);
  int*       pure     = (int*)      alloc((size_t)N * 4);
  int*       vxA      = (int*)      alloc((size_t)N * 4);
  int*       vyA      = (int*)      alloc((size_t)N * 4);
  int*       vzA      = (int*)      alloc((size_t)N * 4);
  int*       keyBL    = (int*)      alloc((size_t)N * 4);
  double*    gc       = (double*)   alloc(64 * 4 * 8);
  float*     tx       = (float*)    alloc((size_t)N * 4);
  float*     ty       = (float*)    alloc((size_t)N * 4);
  float*     tz       = (float*)    alloc((size_t)N * 4);
  int*       missA    = (int*)      alloc((size_t)N * 4);
  int*       bestJ    = (int*)      alloc((size_t)N * 4);
  float*     mag      = (float*)    alloc((size_t)N * 4);
  (void)ws_size; (void)n_in;

  int gb  = (Np2 + THREADS - 1) / THREADS;
  int gbN = (N + THREADS - 1) / THREADS;

  // 1) cluster points by (batch,voxel) key
  k_prep<<<gb, THREADS, 0, stream>>>(grid, batch, label, pBase, pNcls,
                                     key1, idx1, keyBL, N, Np2);
  sort_pairs(key1, idx1, Np2, stream);
  k_scan<<<1, THREADS, 0, stream>>>(key1, idx1, clOfPt, clStart, hdr, N);
  k_stats<<<gb, THREADS, 0, stream>>>(grid, batch, label, idx1, clStart, hdr,
                                      pBase, pNcls, cxA, cyA, czA, bmat,
                                      code, pure, vxA, vyA, vzA, key2, idx2, N, Np2);
  // 2) pure-cluster hash table (sorted FNV keys) for the voxel probe
  sort_pairs(key2, idx2, Np2, stream);
  // 3) code-grouped cluster table for the WMMA fallback
  k_ckey<<<gb, THREADS, 0, stream>>>(code, hdr, key3, idx3, N, Np2);
  sort_pairs(key3, idx3, Np2, stream);
  k_gather<<<gbN, THREADS, 0, stream>>>(idx3, code, bmat, scode, sbmat, N);
  k_initcol<<<1, 128, 0, stream>>>(colStart);
  k_bound<<<gbN, THREADS, 0, stream>>>(scode, colStart, pNcls, pNb, N);
  k_fill<<<1, 1, 0, stream>>>(colStart, hdr, pNcls, pNb);
  // 4) global (batch,label) centers + voxel probe
  k_glob<<<64, THREADS, 0, stream>>>(grid, keyBL, gc, pNcls, pNb, N);
  k_probe<<<gbN, THREADS, 0, stream>>>(grid, batch, label, clOfPt, keyBL,
                                       cxA, cyA, czA, pure, vxA, vyA, vzA,
                                       key2, idx2, gc, tx, ty, tz, missA, N);
  // 5) code-grouped point order, then WMMA nearest-center fallback
  k_pkey<<<gb, THREADS, 0, stream>>>(keyBL, key1, idx1, N, Np2);
  sort_pairs(key1, idx1, Np2, stream);
  int nWaves = (N + 15) / 16;
  k_fallback<<<(nWaves + 7) / 8, 256, 0, stream>>>(grid, keyBL, idx1, sbmat, scode,
                                                   idx3, colStart, bestJ, N);
  // 6) offsets, magnitudes, 99th-percentile threshold, losses
  k_apply<<<gb, THREADS, 0, stream>>>(grid, missA, bestJ, cxA, cyA, czA,
                                      tx, ty, tz, mag, key1, N, Np2);
  sort_pairs(key1, idx1, Np2, stream);
  k_loss<<<1, THREADS, 0, stream>>>(pred, tx, ty, tz, mag, key1, (float*)d_out, N);
}